// HashEmbedder_5007931867219
// MI455X (gfx1250) — compile-verified
//
#include <hip/hip_runtime.h>
#include <cmath>

// HashNeRF / Instant-NGP hash-grid embedding lookup for MI455X (gfx1250).
//
// Perf model: 1 GiB of random 8B gathers into a 64 MB table that fits in the
// 192 MB L2 -> L2-request-rate bound, not FLOP bound. WMMA is inapplicable
// (batched per-point 1x8 x 8x2 matvec, no shared operand). Chip-specific
// levers: NT stores for the 128 MB streaming output so the table stays
// L2-resident; deep load ILP per wave32 to hide L2 gather latency.

#define N_LEVELS 16
#define LOG2_T   19
#define TABLE_SZ (1u << LOG2_T)
#define HASH_MASK (TABLE_SZ - 1u)
#define PRIME_Y  2654435761u
#define PRIME_Z  805459861u
#define BLOCK    256

typedef float v4f __attribute__((ext_vector_type(4)));
typedef float v2f __attribute__((ext_vector_type(2)));

struct ResArr { float v[N_LEVELS]; };

__global__ __launch_bounds__(BLOCK)
void hashgrid_interp_kernel(const float* __restrict__ x,
                            const float* __restrict__ emb,
                            float* __restrict__ out,
                            ResArr res, int npts)
{
    const int n = blockIdx.x * BLOCK + threadIdx.x;
    if (n >= npts) return;

    // Point coords in [0,1)^3. 12B/thread, coalesced enough, tiny traffic.
    const float px = x[3 * n + 0];
    const float py = x[3 * n + 1];
    const float pz = x[3 * n + 2];

    float o[N_LEVELS * 2];

    // Level loop unrolled x4 -> up to 32 independent 8B gathers in flight
    // per wave to hide L2 latency (LOADcnt-limited pipelining).
    #pragma unroll 4
    for (int l = 0; l < N_LEVELS; ++l) {
        const float r  = res.v[l];
        const float sx = px * r, sy = py * r, sz = pz * r;
        const float fx = floorf(sx), fy = floorf(sy), fz = floorf(sz);
        const float wx = sx - fx,   wy = sy - fy,   wz = sz - fz;
        const float ux = 1.0f - wx, uy = 1.0f - wy, uz = 1.0f - wz;

        // uint32 wraparound multiply matches the reference's hash exactly.
        const unsigned bx = (unsigned)fx, by = (unsigned)fy, bz = (unsigned)fz;
        const unsigned hx0 = bx,          hx1 = bx + 1u;
        const unsigned hy0 = by * PRIME_Y, hy1 = hy0 + PRIME_Y;
        const unsigned hz0 = bz * PRIME_Z, hz1 = hz0 + PRIME_Z;

        const v2f* __restrict__ t =
            (const v2f*)emb + (size_t)l * (size_t)TABLE_SZ;

        float ax = 0.0f, ay = 0.0f;
        // 8 corners fully unrolled: 8 independent global_load_b64 gathers.
        #pragma unroll
        for (int c = 0; c < 8; ++c) {
            const unsigned hx = (c & 4) ? hx1 : hx0;
            const unsigned hy = (c & 2) ? hy1 : hy0;
            const unsigned hz = (c & 1) ? hz1 : hz0;
            const unsigned idx = (hx ^ hy ^ hz) & HASH_MASK;

            const float wcx = (c & 4) ? wx : ux;
            const float wcy = (c & 2) ? wy : uy;
            const float wcz = (c & 1) ? wz : uz;
            const float wc  = wcx * wcy * wcz;

            const v2f e = t[idx];            // RT hint: keep table in L2
            ax = fmaf(wc, e.x, ax);
            ay = fmaf(wc, e.y, ay);
        }
        o[2 * l + 0] = ax;
        o[2 * l + 1] = ay;
    }

    // 32 floats per point, [N, 32] layout. Non-temporal b128 stores: the
    // 128 MB output stream must NOT evict the 64 MB embedding table from L2.
    v4f* __restrict__ op = (v4f*)(out + (size_t)n * 32);
    #pragma unroll
    for (int i = 0; i < 8; ++i) {
        v4f v = { o[4 * i + 0], o[4 * i + 1], o[4 * i + 2], o[4 * i + 3] };
        __builtin_nontemporal_store(v, op + i);
    }
}

extern "C" void kernel_launch(void* const* d_in, const int* in_sizes, int n_in,
                              void* d_out, int out_size, void* d_ws, size_t ws_size,
                              hipStream_t stream) {
    const float* x   = (const float*)d_in[0];   // [N, 3] f32
    const float* emb = (const float*)d_in[1];   // [16, 2^19, 2] f32
    float* out       = (float*)d_out;           // [N, 32] f32

    const int npts = in_sizes[0] / 3;

    // Replicate numpy's float64 resolution computation bit-for-bit intent:
    // b = exp((ln 512 - ln 16)/15); res_l = floor(16 * b**l), then -> f32.
    // Done in double on host to avoid off-by-one floors at l=3,6,9,12,15
    // where 16*b^l is within 1 ulp of an integer.
    ResArr r;
    const double b = std::exp((std::log(512.0) - std::log(16.0)) / 15.0);
    for (int l = 0; l < N_LEVELS; ++l) {
        r.v[l] = (float)std::floor(16.0 * std::pow(b, (double)l));
    }

    const int grid = (npts + BLOCK - 1) / BLOCK;
    hashgrid_interp_kernel<<<grid, BLOCK, 0, stream>>>(x, emb, out, r, npts);
}